// QuantizedLinearTriton_12360915878148
// MI455X (gfx1250) — compile-verified
//
#include <hip/hip_runtime.h>

typedef __attribute__((ext_vector_type(2))) float v2f;
typedef __attribute__((ext_vector_type(4))) float v4f;
typedef __attribute__((ext_vector_type(8))) float v8f;
typedef __attribute__((ext_vector_type(4))) int   v4i;

#define AS1 __attribute__((address_space(1)))
#define AS3 __attribute__((address_space(3)))

#ifndef __has_builtin
#define __has_builtin(x) 0
#endif

#if __has_builtin(__builtin_amdgcn_global_load_async_to_lds_b128)
#define QL_ASYNC 1
#else
#define QL_ASYNC 0
#endif

constexpr int BM  = 128;        // block rows of x
constexpr int BN  = 128;        // block cols (out features)
constexpr int BK  = 16;         // K chunk
constexpr int AS  = BK + 4;     // 20 floats: A row stride (conflict-free, 16B aligned rows)
constexpr int PS  = 2 * BN + 32;// 288 floats: B pair-row stride (mod 64 = 32 -> disjoint bank halves)
constexpr int NTH = 256;        // 8 wave32s

__device__ __forceinline__ void async_wait_all() {
#if QL_ASYNC
#if __has_builtin(__builtin_amdgcn_s_wait_asynccnt)
  __builtin_amdgcn_s_wait_asynccnt(0);
#else
  asm volatile("s_wait_asynccnt 0" ::: "memory");
#endif
#endif
}

// Copy a BM x BK f32 tile of x into LDS (async-to-LDS when available).
__device__ __forceinline__ void fill_a(const float* __restrict__ x, int K,
                                       int rowBase, int kBase,
                                       float* __restrict__ sAbuf, int t) {
#pragma unroll
  for (int j = 0; j < 2; ++j) {
    int seg = t * 2 + j;            // 0..511 : 16-byte segments
    int row = seg >> 2;             // 0..127
    int sc  = (seg & 3) * 4;        // 0,4,8,12
    const float* g = x + (size_t)(rowBase + row) * K + kBase + sc;
    float* l = sAbuf + row * AS + sc;
#if QL_ASYNC
    __builtin_amdgcn_global_load_async_to_lds_b128(
        (AS1 v4i*)g, (AS3 v4i*)l, 0, 0);
#else
    *(v4f*)l = *(const v4f*)g;
#endif
  }
}

// Dequantize a BN x BK tile of W into LDS in pair-interleaved B layout:
//   sB[kp * PS + 2*n + parity] = W[n][kBase + 2*kp + parity]
// weight_packed[n, K/2] int32, one byte per element: hi nibble -> k even, lo -> k odd.
__device__ __forceinline__ void fill_b(const int* __restrict__ wp,
                                       const float* __restrict__ scales,
                                       const float* __restrict__ sCent,
                                       float* __restrict__ sBbuf,
                                       int K, int nBase, int kBase, int t) {
  int n  = t >> 1;                  // 0..127
  int p0 = (t & 1) * 4;             // packed-int col within chunk: 0 or 4
  const int* gw = wp + (size_t)(nBase + n) * (K >> 1) + (kBase >> 1) + p0;
  v4i pk = *(const v4i*)gw;         // 4 packed bytes-in-int32, coalesced 16B loads
  float sc = scales[(size_t)(nBase + n) * (K >> 6) + (kBase >> 6)]; // block of 64 k
  int vals[4] = {pk.x, pk.y, pk.z, pk.w};
#pragma unroll
  for (int j = 0; j < 4; ++j) {
    int v = vals[j];
    int p = p0 + j;                 // pair index 0..7 -> k pair (2p, 2p+1)
    v2f w2;
    w2.x = sCent[(v >> 4) & 15] * sc;   // k = 2p   (hi nibble)
    w2.y = sCent[v & 15] * sc;          // k = 2p+1 (lo nibble)
    *(v2f*)(sBbuf + p * PS + 2 * n) = w2;  // one aligned b64 store
  }
}

// One BK chunk of WMMA f32 16x16x4 accumulation. Wave owns 32x64 (2x4 tiles).
__device__ __forceinline__ void compute_chunk(const float* __restrict__ sAbuf,
                                              const float* __restrict__ sBbuf,
                                              v8f acc[2][4], int lane, int wm, int wn) {
  const int l15  = lane & 15;
  const int lhi  = lane >> 4;          // 0: K=0,1 ; 1: K=2,3
  const int koff = lhi << 1;
#pragma unroll
  for (int k4 = 0; k4 < BK; k4 += 4) {
    v2f a[2];
#pragma unroll
    for (int mi = 0; mi < 2; ++mi) {
      const float* ap = sAbuf + (wm * 32 + mi * 16 + l15) * AS + k4 + koff;
      a[mi] = *(const v2f*)ap;         // aligned b64; all 64 banks hit exactly once
    }
    const int kp = (k4 >> 1) + lhi;    // pair row for this half-wave
    v2f b[4];
#pragma unroll
    for (int ni = 0; ni < 4; ++ni) {
      const float* bp = sBbuf + kp * PS + 2 * (wn * 64 + ni * 16 + l15);
      b[ni] = *(const v2f*)bp;         // single aligned b64, conflict-free
    }
#pragma unroll
    for (int mi = 0; mi < 2; ++mi)
#pragma unroll
      for (int ni = 0; ni < 4; ++ni)
        acc[mi][ni] = __builtin_amdgcn_wmma_f32_16x16x4_f32(
            false, a[mi], false, b[ni], (short)0, acc[mi][ni], false, false);
  }
}

__global__ __launch_bounds__(NTH) void qlinear_wmma_f32(
    const float* __restrict__ x, const int* __restrict__ wp,
    const float* __restrict__ cent, const float* __restrict__ scales,
    float* __restrict__ out, int M, int N, int K) {
  __shared__ float sA[2][BM * AS];          // 20.0 KB
  __shared__ float sB[2][(BK / 2) * PS];    // 18.0 KB
  __shared__ float sCent[16];

  const int t    = threadIdx.x;
  const int lane = t & 31;
  const int wave = t >> 5;
  const int wm   = wave & 3;             // 4 waves down (32 rows each)
  const int wn   = wave >> 2;            // 2 waves across (64 cols each)
  const int rowBase = blockIdx.y * BM;
  const int nBase   = blockIdx.x * BN;

  if (t < 16) sCent[t] = cent[t];
  __syncthreads();                       // centroids visible before any dequant

  v8f acc[2][4];
  const v8f z = {0.f, 0.f, 0.f, 0.f, 0.f, 0.f, 0.f, 0.f};
#pragma unroll
  for (int mi = 0; mi < 2; ++mi)
#pragma unroll
    for (int ni = 0; ni < 4; ++ni) acc[mi][ni] = z;

  // Prologue: stage 0
  fill_a(x, K, rowBase, 0, sA[0], t);
  fill_b(wp, scales, sCent, sB[0], K, nBase, 0, t);
  async_wait_all();
  __syncthreads();

  const int NC = K / BK;
  for (int c = 0; c < NC; ++c) {
    const int cur = c & 1;
    if (c + 1 < NC) {                    // issue next stage while computing current
      fill_a(x, K, rowBase, (c + 1) * BK, sA[cur ^ 1], t);
      fill_b(wp, scales, sCent, sB[cur ^ 1], K, nBase, (c + 1) * BK, t);
    }
    compute_chunk(sA[cur], sB[cur], acc, lane, wm, wn);
    async_wait_all();                    // drain ASYNCcnt before the barrier
    __syncthreads();
  }

  // Epilogue: C/D layout -> VGPR v is M=v (lanes 0-15) / M=v+8 (lanes 16-31)
  const int cTop = nBase + wn * 64;
  const int rTop = rowBase + wm * 32 + ((lane >> 4) << 3);
  const int l15  = lane & 15;
#pragma unroll
  for (int mi = 0; mi < 2; ++mi) {
#pragma unroll
    for (int ni = 0; ni < 4; ++ni) {
      const int col = cTop + ni * 16 + l15;
#pragma unroll
      for (int v = 0; v < 8; ++v) {
        const int row = rTop + mi * 16 + v;
        out[(size_t)row * N + col] = acc[mi][ni][v];
      }
    }
  }
}

extern "C" void kernel_launch(void* const* d_in, const int* in_sizes, int n_in,
                              void* d_out, int out_size, void* d_ws, size_t ws_size,
                              hipStream_t stream) {
  const float* x      = (const float*)d_in[0];
  const int*   wp     = (const int*)d_in[1];
  const float* cent   = (const float*)d_in[2];
  const float* scales = (const float*)d_in[3];
  float*       out    = (float*)d_out;

  const int K = 4096;                                    // in features (fixed by reference)
  const int M = in_sizes[0] / K;                         // 8192
  const int N = (int)(((long long)in_sizes[1] * 2) / K); // 4096

  dim3 grid(N / BN, M / BM);
  dim3 block(NTH);
  hipLaunchKernelGGL(qlinear_wmma_f32, grid, block, 0, stream,
                     x, wp, cent, scales, out, M, N, K);
}